// Direct_Conv2d_6854767804819
// MI455X (gfx1250) — compile-verified
//
#include <hip/hip_runtime.h>
#include <hip/hip_bf16.h>

typedef __attribute__((ext_vector_type(2))) float v2f;
typedef __attribute__((ext_vector_type(8))) float v8f;

// Direct conv2d as implicit GEMM on V_WMMA_F32_16X16X4_F32.
// Per tap (r,s):  D(16x16) += A(16x4) x B(4x16)
//   A: M = output channel (8 real, 8 zero), K = input channel (C==4 == WMMA K)
//   B: K = input channel, N = 16 consecutive output x positions
// Output D: VGPR i holds channel i for lanes 0..15 (N = pixel) -> coalesced stores.
__global__ __launch_bounds__(256) void Direct_Conv2d_6854767804819_kernel(
    const float* __restrict__ in,    // (16, 4, 512, 512)
    const float* __restrict__ filt,  // (8, 4, 3, 3)
    float* __restrict__ out)         // (16, 8, 510, 510)
{
    constexpr int Hin = 512, Win = 512, Cin = 4;
    constexpr int Hout = 510, Wout = 510, Cout = 8;
    constexpr int XT = 32;  // ceil(510/16) tiles along W

    const int lane = threadIdx.x & 31;
    const int wave = threadIdx.x >> 5;

    int tile = blockIdx.x * 8 + wave;        // 261120 tiles total, exact
    const int xt = tile & (XT - 1);
    int t = tile >> 5;                       // = n*510 + y
    const int y = t % Hout;
    const int n = t / Hout;
    const int x0 = xt << 4;

    const int col = lane & 15;               // A: M (out channel); B: N (pixel)
    const int c0  = (lane >> 4) << 1;        // K base for this half-wave: 0 or 2

    // ---- Preload filter A-fragments for all 9 taps ----
    // afrag[r][s] = { filt[o][c0][r][s], filt[o][c0+1][r][s] }, o = col (zero if o>=8)
    v2f afrag[3][3];
    {
        const int o = col;
        const bool valid = (o < Cout);
        const float* fb = filt + ((size_t)o * Cin + c0) * 9;  // 3*3 = 9
#pragma unroll
        for (int r = 0; r < 3; ++r) {
#pragma unroll
            for (int s = 0; s < 3; ++s) {
                const int rs = r * 3 + s;
                v2f a;
                a.x = valid ? fb[rs]     : 0.0f;
                a.y = valid ? fb[9 + rs] : 0.0f;  // next input channel: stride R*S=9
                afrag[r][s] = a;
            }
        }
    }

    // ---- Accumulate 9 taps via WMMA ----
    v8f acc = {};
    const float* inb = in + (size_t)n * Cin * Hin * Win + (size_t)c0 * Hin * Win;
    constexpr size_t cstride = (size_t)Hin * Win;

#pragma unroll
    for (int r = 0; r < 3; ++r) {
        const int yy = y + r;                // <= 511, always in range
#pragma unroll
        for (int s = 0; s < 3; ++s) {
            int x = x0 + col + s;
            if (x > Win - 1) x = Win - 1;    // clamp: only feeds masked-out columns
            const float* p = inb + (size_t)yy * Win + x;
            v2f b;
            b.x = p[0];                      // input channel c0
            b.y = p[cstride];                // input channel c0+1
            // (neg_a, A, neg_b, B, c_mod, C, reuse_a, reuse_b)
            acc = __builtin_amdgcn_wmma_f32_16x16x4_f32(
                false, afrag[r][s], false, b, (short)0, acc, false, false);
        }
    }

    // ---- Store: VGPR i = channel i, lanes 0..15 = 16 consecutive x ----
    if (lane < 16) {
        const int xo = x0 + col;
        if (xo < Wout) {
            float* op = out + (((size_t)n * Cout) * Hout + y) * Wout + xo;
            constexpr size_t ostride = (size_t)Hout * Wout;
#pragma unroll
            for (int i = 0; i < 8; ++i)
                op[(size_t)i * ostride] = acc[i];
        }
    }
}

extern "C" void kernel_launch(void* const* d_in, const int* in_sizes, int n_in,
                              void* d_out, int out_size, void* d_ws, size_t ws_size,
                              hipStream_t stream) {
    const float* in   = (const float*)d_in[0];
    const float* filt = (const float*)d_in[1];
    float* out        = (float*)d_out;

    // tiles = 16 (n) * 510 (y) * 32 (x-tiles) = 261120 waves; 8 waves/block
    const int blocks = 261120 / 8;  // 32640
    Direct_Conv2d_6854767804819_kernel<<<blocks, 256, 0, stream>>>(in, filt, out);
}